// AttentionPooling_67173288509680
// MI455X (gfx1250) — compile-verified
//
#include <hip/hip_runtime.h>
#include <hip/hip_bf16.h>
#include <math.h>

// Problem constants from the reference
#define BATCH 64
#define NNODES 600
#define CHANS 256
#define THRESHOLD 0.003f
#define EPSV 1e-7f

typedef float v2f __attribute__((ext_vector_type(2)));
typedef float v8f __attribute__((ext_vector_type(8)));

// ---------------------------------------------------------------------------
// K1: scores[r] = dot(x[r,:], W[:])  for r in [0, B*N), via V_WMMA_F32_16X16X4_F32.
// One wave32 per 16-row tile; 64 accumulating WMMAs over C=256 (K=4 chunks).
// A-frag (16x4 f32, ISA 7.12.2): lane L holds M = L%16, K = 2*(L>>4)+vgpr.
// B-frag: W replicated across all 16 N columns -> every D column equals the
// score vector, so the result is independent of the B/D N-lane mapping detail.
// C/D layout: lanes 0..15 hold M=0..7 in VGPR 0..7; lanes 16..31 hold M=8..15.
// ---------------------------------------------------------------------------
__global__ __launch_bounds__(32) void scores_wmma_kernel(
    const float* __restrict__ x, const float* __restrict__ W,
    float* __restrict__ scores) {
  const int lane  = threadIdx.x;            // 0..31
  const int tile  = blockIdx.x;             // 16 rows per tile
  const int row   = tile * 16 + (lane & 15);
  const int khalf = (lane >> 4) * 2;        // 0 for lanes 0-15, 2 for 16-31
  const float* __restrict__ xrow = x + (size_t)row * CHANS;

  v8f c = {0.f, 0.f, 0.f, 0.f, 0.f, 0.f, 0.f, 0.f};
  for (int k0 = 0; k0 < CHANS; k0 += 4) {
    v2f a, b;
    a.x = xrow[k0 + khalf];
    a.y = xrow[k0 + khalf + 1];
    b.x = W[k0 + khalf];        // same for every N column of B
    b.y = W[k0 + khalf + 1];
    // 8 args: (neg_a, A, neg_b, B, c_mod, C, reuse_a, reuse_b)
    c = __builtin_amdgcn_wmma_f32_16x16x4_f32(false, a, false, b,
                                              (short)0, c, false, false);
  }
  if (lane == 0) {
    #pragma unroll
    for (int m = 0; m < 8; ++m) scores[tile * 16 + m] = c[m];
  } else if (lane == 16) {
    #pragma unroll
    for (int m = 0; m < 8; ++m) scores[tile * 16 + 8 + m] = c[m];
  }
}

// ---------------------------------------------------------------------------
// K2: one block per batch (640 threads >= N=600).
//  - e = exp(score)*mask; denom = sum_n e  (LDS tree reduction)
//  - alpha = e/(denom+eps); keep = mask && alpha > THRESHOLD
//  - stable partition via inclusive Hillis-Steele scan of keep:
//      kept n  -> pos = (#keep <= n) - 1
//      drop n  -> pos = M + n - (#keep <= n)
//    perm[pos] = n  (matches jax.lax.top_k stable tie-break on 0/1 values)
//  - mask_k output: 1.0 for j < M else 0.0
// ---------------------------------------------------------------------------
#define SCAN_T 640
__global__ __launch_bounds__(SCAN_T) void alpha_scan_kernel(
    const float* __restrict__ scores, const unsigned char* __restrict__ mask,
    float* __restrict__ alpha, int* __restrict__ perm, int* __restrict__ Marr,
    float* __restrict__ maskk_out) {
  __shared__ float sf[SCAN_T];
  __shared__ int   si[SCAN_T];
  const int b   = blockIdx.x;
  const int tid = threadIdx.x;
  const int base = b * NNODES;

  const bool in = (tid < NNODES);
  const bool m  = in && (mask[base + tid] != 0);
  const float e = m ? __expf(scores[base + tid]) : 0.f;

  // denom reduction
  sf[tid] = e;
  __syncthreads();
  for (int stride = 512; stride > 0; stride >>= 1) {
    if (tid < stride && tid + stride < SCAN_T) sf[tid] += sf[tid + stride];
    __syncthreads();
  }
  const float denom = sf[0];

  const float a = e / (denom + EPSV);
  const int keep = (m && a > THRESHOLD) ? 1 : 0;
  if (in) alpha[base + tid] = a;

  // inclusive scan of keep over all 640 slots (keep==0 for tid>=600)
  si[tid] = keep;
  __syncthreads();
  for (int off = 1; off < SCAN_T; off <<= 1) {
    int t = si[tid];
    if (tid >= off) t += si[tid - off];
    __syncthreads();
    si[tid] = t;
    __syncthreads();
  }
  const int kc = si[tid];          // #keep in [0..tid]
  const int M  = si[SCAN_T - 1];   // total kept in this batch

  if (in) {
    const int pos = keep ? (kc - 1) : (M + tid - kc);
    perm[base + pos] = tid;
    maskk_out[base + tid] = (tid < M) ? 1.f : 0.f;
  }
  if (tid == 0) Marr[b] = M;
}

// ---------------------------------------------------------------------------
// K3: x_out[b,j,:] = x[b,perm[j],:] * alpha[b,perm[j]]   (ALL rows, no mask —
// the reference gathers x without re-masking). One block = one output row,
// 256 threads = C channels, fully coalesced in and out.
// ---------------------------------------------------------------------------
__global__ __launch_bounds__(CHANS) void x_gather_kernel(
    const float* __restrict__ x, const float* __restrict__ alpha,
    const int* __restrict__ perm, float* __restrict__ xout) {
  const int j = blockIdx.x;
  const int b = blockIdx.y;
  const int c = threadIdx.x;
  const int p = perm[b * NNODES + j];
  const float a = alpha[b * NNODES + p];
  xout[((size_t)b * NNODES + j) * CHANS + c] =
      x[((size_t)b * NNODES + p) * CHANS + c] * a;
}

// ---------------------------------------------------------------------------
// K4: A_out[b,i,j] = A[b,perm[i],perm[j]] inside the MxM kept block, else 0.
// perm is ascending within each partition segment, so the column gather is
// nearly coalesced. Writes are fully coalesced.
// ---------------------------------------------------------------------------
__global__ __launch_bounds__(256) void a_gather_kernel(
    const float* __restrict__ A, const int* __restrict__ perm,
    const int* __restrict__ Marr, float* __restrict__ Aout) {
  const int j = blockIdx.x * 256 + threadIdx.x;
  const int i = blockIdx.y;
  const int b = blockIdx.z;
  if (j >= NNODES) return;
  const int M = Marr[b];
  float v = 0.f;
  if (i < M && j < M) {
    const int pi = perm[b * NNODES + i];
    const int pj = perm[b * NNODES + j];
    v = A[((size_t)b * NNODES + pi) * NNODES + pj];
  }
  Aout[((size_t)b * NNODES + i) * NNODES + j] = v;
}

// ---------------------------------------------------------------------------
// Launch: inputs (setup_inputs order): x (f32 B*N*C), A (f32 B*N*N),
// mask (bool, 1 byte/elem), W (f32 1*C).
// d_out (float): [x_out | A_out | mask_k] concatenated flat.
// ---------------------------------------------------------------------------
extern "C" void kernel_launch(void* const* d_in, const int* in_sizes, int n_in,
                              void* d_out, int out_size, void* d_ws, size_t ws_size,
                              hipStream_t stream) {
  const float*         x    = (const float*)d_in[0];
  const float*         A    = (const float*)d_in[1];
  const unsigned char* mask = (const unsigned char*)d_in[2];
  const float*         W    = (const float*)d_in[3];

  // workspace carve-out
  float* scores = (float*)d_ws;                 // B*N
  float* alpha  = scores + BATCH * NNODES;      // B*N
  int*   perm   = (int*)(alpha + BATCH * NNODES); // B*N
  int*   Marr   = perm + BATCH * NNODES;        // B

  float* out    = (float*)d_out;
  float* xout   = out;                                          // B*N*C
  float* Aout   = out + (size_t)BATCH * NNODES * CHANS;         // B*N*N
  float* maskk  = Aout + (size_t)BATCH * NNODES * NNODES;       // B*N

  // K1: GEMV on the matrix pipe (wmma f32 16x16x4), 2400 tiles of 16 rows
  scores_wmma_kernel<<<(BATCH * NNODES) / 16, 32, 0, stream>>>(x, W, scores);

  // K2: per-batch normalize + stable-partition scan
  alpha_scan_kernel<<<BATCH, SCAN_T, 0, stream>>>(scores, mask, alpha, perm,
                                                  Marr, maskk);

  // K3: x gather (scaled rows, all 600 positions)
  x_gather_kernel<<<dim3(NNODES, BATCH), CHANS, 0, stream>>>(x, alpha, perm, xout);

  // K4: A gather, zero outside MxM
  a_gather_kernel<<<dim3((NNODES + 255) / 256, NNODES, BATCH), 256, 0, stream>>>(
      A, perm, Marr, Aout);
}